// ConnectedFilterLayer_29884382446231
// MI455X (gfx1250) — compile-verified
//
#include <hip/hip_runtime.h>
#include <hip/hip_bf16.h>

typedef __attribute__((ext_vector_type(2))) float v2f;
typedef __attribute__((ext_vector_type(8))) float v8f;

#define NB    4
#define NC    3
#define NPAIR 12          // B*C
#define NN    262144      // nodes per (b,c)
#define NP    262145      // nodes + sentinel
#define NHW   1048576     // H*W
#define NG    3           // groups

// ---------------------------------------------------------------------------
// K0: copy parent -> ptr0 (with sentinel self-loop at node N), zero the pad
// channel (w) of acc0 and the whole sentinel row. acc layout: float4 per node,
// (contrib_g0, contrib_g1, contrib_g2, 0), pair stride = NP float4s.
// ---------------------------------------------------------------------------
__global__ void cf_init_kernel(const int* __restrict__ parent,
                               float4* __restrict__ acc0,
                               int* __restrict__ ptr0) {
    size_t t = (size_t)blockIdx.x * blockDim.x + threadIdx.x;
    const size_t TOT = (size_t)NPAIR * NP;
    if (t >= TOT) return;
    int p = (int)(t / NP);
    int i = (int)(t - (size_t)p * NP);
    if (i < NN) {
        ptr0[t] = parent[(size_t)p * NN + i];
        acc0[t].w = 0.0f;
    } else {                           // sentinel row: acc = 0, ptr = self
        ptr0[t] = NN;
        acc0[t] = make_float4(0.f, 0.f, 0.f, 0.f);
    }
}

// ---------------------------------------------------------------------------
// K1: per 16-node tile, compute all 3 group scores with one
// V_WMMA_F32_16X16X4_F32:  D(16x16) = A(16x4) * Bw(4x16)
//   A row m = [attr0, attr1, attr2, 1.0] of node m
//   Bw col g = [W over attrs ; bias_g],  cols 3..15 = 0 (unused)
// A layout (32-bit 16x4): lanes 0-15 hold K=0(v0),K=1(v1); lanes 16-31 hold
// K=2(v0),K=3(v1).  B layout (4x16) mirrored: lanes 0-15 hold rows K=0/K=1,
// lanes 16-31 hold rows K=2/K=3, lane index = column N.
// D layout: VGPR r = row M=r (lanes 0-15) / M=r+8 (lanes 16-31), lane = col N.
// ---------------------------------------------------------------------------
__global__ void cf_score_wmma_kernel(const float* __restrict__ attrs,
                                     const float* __restrict__ residue,
                                     const float* __restrict__ W0,
                                     const float* __restrict__ W1,
                                     const float* __restrict__ W2,
                                     const float* __restrict__ b0,
                                     const float* __restrict__ b1,
                                     const float* __restrict__ b2,
                                     float* __restrict__ acc0) {
    const int wave  = blockIdx.x * (blockDim.x >> 5) + (threadIdx.x >> 5);
    const int lane  = threadIdx.x & 31;
    const int TPP   = NN / 16;                 // tiles per pair
    const int pair  = wave / TPP;
    const int tnode = (wave - pair * TPP) * 16;
    const int m     = lane & 15;
    const bool hi   = lane >= 16;

    // ---- weights / biases: uniform loads, hoisted out of lane predicates
    const float w0  = W0[0];
    const float w10 = W1[0];
    const float w11 = W1[1];
    const float w2  = W2[0];
    const float bb0 = b0[0];
    const float bb1 = b1[0];
    const float bb2 = b2[0];

    // ---- A fragment
    const float* ap = attrs + ((size_t)pair * NN + (size_t)(tnode + m)) * 3;
    v2f a;
    if (!hi) { a.x = ap[0]; a.y = ap[1]; }
    else     { a.x = ap[2]; a.y = 1.0f;  }

    // ---- B fragment (weights + bias), column = group, row = feature/bias
    float v0, v1;
    if (!hi) {                  // K=0 row in v0, K=1 row in v1
        v0 = (m == 0) ? w0  : 0.0f;
        v1 = (m == 1) ? w10 : ((m == 2) ? w2 : 0.0f);
    } else {                    // K=2 row in v0, K=3 (bias) row in v1
        v0 = (m == 1) ? w11 : 0.0f;
        v1 = (m == 0) ? bb0 : ((m == 1) ? bb1 : ((m == 2) ? bb2 : 0.0f));
    }
    v2f b; b.x = v0; b.y = v1;

    v8f c = {};
    v8f d = __builtin_amdgcn_wmma_f32_16x16x4_f32(
        /*neg_a=*/false, a, /*neg_b=*/false, b,
        /*c_mod=*/(short)0, c, /*reuse_a=*/false, /*reuse_b=*/false);

    // ---- sigmoid (v_exp_f32 + v_rcp_f32), multiply by residue, scatter into
    //      packed acc (g channel of float4)
    if (m < NG) {
        const int mb = hi ? 8 : 0;
        const int g  = m;
        float* outp       = acc0 + ((size_t)pair * NP + (size_t)(tnode + mb)) * 4 + g;
        const float* resp = residue + (size_t)pair * NN + (size_t)(tnode + mb);
#pragma unroll
        for (int r = 0; r < 8; ++r) {
            float s = __builtin_amdgcn_rcpf(1.0f + __expf(-d[r]));
            outp[(size_t)r * 4] = s * resp[r];
        }
    }
}

// ---------------------------------------------------------------------------
// K2: one pointer-jumping round.
//   accD[i] = accS[i] + accS[ptrS[i]];  ptrD[i] = ptrS[ptrS[i]]
// All b128/b32 global traffic; per-pair working set (~4.2 MB) lives in L2.
// ---------------------------------------------------------------------------
__global__ void cf_jump_kernel(const float4* __restrict__ accS,
                               const int* __restrict__ ptrS,
                               float4* __restrict__ accD,
                               int* __restrict__ ptrD) {
    size_t t = (size_t)blockIdx.x * blockDim.x + threadIdx.x;
    const size_t TOT = (size_t)NPAIR * NP;
    if (t >= TOT) return;
    int p = (int)(t / NP);
    size_t base = (size_t)p * NP;
    int j = ptrS[t];
    float4 ai = accS[t];
    float4 aj = accS[base + (size_t)j];
    float4 r;
    r.x = ai.x + aj.x;
    r.y = ai.y + aj.y;
    r.z = ai.z + aj.z;
    r.w = 0.0f;
    accD[t] = r;
    ptrD[t] = ptrS[base + (size_t)j];
}

// ---------------------------------------------------------------------------
// K3: gather f[pixel_node] for all 3 groups; 4 pixels per thread so the
// pixel_node read and all 3 plane writes are b128.
// out index = ((b*C + c)*3 + g)*HW + pix  =  (p*3 + g)*HW + pix
// ---------------------------------------------------------------------------
__global__ void cf_gather_kernel(const int* __restrict__ pixel_node,
                                 const float4* __restrict__ acc,
                                 float* __restrict__ out) {
    size_t t = (size_t)blockIdx.x * blockDim.x + threadIdx.x;
    const size_t QP = NHW / 4;                 // pixel quads per pair
    const size_t TOT = (size_t)NPAIR * QP;
    if (t >= TOT) return;
    int p   = (int)(t / QP);
    size_t q = t - (size_t)p * QP;

    const int4 idx = ((const int4*)pixel_node)[t];   // flat layout matches t
    const float4* accp = acc + (size_t)p * NP;
    float4 f0 = accp[idx.x];
    float4 f1 = accp[idx.y];
    float4 f2 = accp[idx.z];
    float4 f3 = accp[idx.w];

    float* base = out + (size_t)p * NG * NHW + q * 4;
    ((float4*)(base))[0]                    = make_float4(f0.x, f1.x, f2.x, f3.x);
    ((float4*)(base + (size_t)NHW))[0]      = make_float4(f0.y, f1.y, f2.y, f3.y);
    ((float4*)(base + (size_t)2 * NHW))[0]  = make_float4(f0.z, f1.z, f2.z, f3.z);
}

// ---------------------------------------------------------------------------
extern "C" void kernel_launch(void* const* d_in, const int* in_sizes, int n_in,
                              void* d_out, int out_size, void* d_ws, size_t ws_size,
                              hipStream_t stream) {
    const float* attrs      = (const float*)d_in[0];
    const float* residue    = (const float*)d_in[1];
    const float* W0         = (const float*)d_in[2];
    const float* W1         = (const float*)d_in[3];
    const float* W2         = (const float*)d_in[4];
    const float* b0         = (const float*)d_in[5];
    const float* b1         = (const float*)d_in[6];
    const float* b2         = (const float*)d_in[7];
    const int*   parent     = (const int*)d_in[8];
    const int*   pixel_node = (const int*)d_in[9];
    float*       out        = (float*)d_out;

    // Workspace: acc0 | acc1 (float4 x NPAIR*NP each) | ptr0 | ptr1
    float4* acc0 = (float4*)d_ws;
    float4* acc1 = acc0 + (size_t)NPAIR * NP;
    int*    ptr0 = (int*)(acc1 + (size_t)NPAIR * NP);
    int*    ptr1 = ptr0 + (size_t)NPAIR * NP;

    const size_t TOT_N = (size_t)NPAIR * NP;
    const int blk = 256;

    // K0: init ptr0 / pad / sentinel
    cf_init_kernel<<<(unsigned)((TOT_N + blk - 1) / blk), blk, 0, stream>>>(
        parent, acc0, ptr0);

    // K1: WMMA score kernel — one wave per 16-node tile, 8 waves per block
    {
        int waves  = NPAIR * (NN / 16);        // 196608
        int blocks = waves / 8;                // 24576
        cf_score_wmma_kernel<<<blocks, blk, 0, stream>>>(
            attrs, residue, W0, W1, W2, b0, b1, b2, (float*)acc0);
    }

    // K2: 18 pointer-jumping rounds; even count ends with result in acc0
    {
        unsigned g = (unsigned)((TOT_N + blk - 1) / blk);
        for (int r = 0; r < 18; ++r) {
            if (r & 1)
                cf_jump_kernel<<<g, blk, 0, stream>>>(acc1, ptr1, acc0, ptr0);
            else
                cf_jump_kernel<<<g, blk, 0, stream>>>(acc0, ptr0, acc1, ptr1);
        }
    }

    // K3: gather to output planes
    {
        const size_t TOT3 = (size_t)NPAIR * (NHW / 4);
        cf_gather_kernel<<<(unsigned)((TOT3 + blk - 1) / blk), blk, 0, stream>>>(
            pixel_node, acc0, out);
    }
}